// FaradayLawLoss_53790170415189
// MI455X (gfx1250) — compile-verified
//
#include <hip/hip_runtime.h>

// ---------------------------------------------------------------------------
// Faraday-law residual loss:  out[n,d,h,w] = (1/3) * sum_c |curl(E) + i*wu*H|^2
// Memory-bound stencil (~0.3 FLOP/B): target is the 23.3 TB/s HBM roofline
// (~218 MB/pass => ~9.4 us floor). CDNA5 paths used:
//   * global_load_async_to_lds_b128 + s_wait_asynccnt (ASYNCcnt DMA) to stage
//     the 4x-reused plane-d E-field tile into LDS (zeroed halos give the
//     conv's zero padding for free),
//   * non-temporal loads/stores for the use-once H-field and output so the
//     reused E planes keep the 192MB L2 to themselves.
// WMMA intentionally NOT used: arithmetic intensity is ~40x below the matrix
// crossover; any matmul formulation inflates FLOPs with zero byte savings.
// ---------------------------------------------------------------------------

#define NB     2
#define DIM    128
#define PLANE  (DIM * DIM)        // 16384
#define VOL    (DIM * DIM * DIM)  // 2097152 (channel stride)
#define HT     4                  // output h-rows per block
#define ROWS   (HT + 2)           // staged rows incl. halo
#define ECHS   6                  // e channels staged (e_re 0-2, e_im 3-5)
#define RSTR   136                // floats per staged row: 4 pad | 128 | 1 | 3 pad
#define LDSF   (ECHS * ROWS * RSTR)  // 4896 floats = 19584 B

__device__ __forceinline__ unsigned lds_addr32(const void* p) {
  // low 32 bits of a generic shared pointer == LDS byte offset
  return (unsigned)(unsigned long long)p;
}

__global__ __launch_bounds__(256) void faraday_loss_kernel(
    const float* __restrict__ f, float* __restrict__ out) {
  __shared__ __align__(16) float els[LDSF];

  const int tid = threadIdx.x;
  const int bid = blockIdx.x;
  const int ht  = bid & 31;          // 32 h-tiles
  const int d   = (bid >> 5) & 127;  // depth plane
  const int n   = bid >> 12;         // batch
  const int h0  = ht * HT;

  // ---- 1) zero-fill LDS (provides conv zero-padding halos) ----
  float4* els4 = (float4*)els;
  for (int i = tid; i < LDSF / 4; i += 256)
    els4[i] = make_float4(0.f, 0.f, 0.f, 0.f);
  __syncthreads();  // ds stores retire before async DMA writes can overlap

  // ---- 2) async DMA: stage E (ch 0..5), plane d, rows h0-1..h0+HT ----
  // one b128 per lane => one 128-float row per wave-instruction
  const int lane = tid & 31;
  const int wv   = tid >> 5;  // 8 waves
  for (int p = wv; p < ECHS * ROWS; p += 8) {  // wave-uniform trip & predicate
    const int ch = p / ROWS;
    const int r  = p % ROWS;
    const int gh = h0 + r - 1;
    if (gh >= 0 && gh < DIM) {
      const float* gsrc =
          f + ((size_t)(n * 12 + ch) * VOL + (size_t)d * PLANE +
               (size_t)gh * DIM + lane * 4);
      unsigned loff = lds_addr32(&els[(ch * ROWS + r) * RSTR + 4 + lane * 4]);
      asm volatile("global_load_async_to_lds_b128 %0, %1, off"
                   :: "v"(loff), "v"(gsrc)
                   : "memory");
    }
  }
  asm volatile("s_wait_asynccnt 0" ::: "memory");
  __syncthreads();

  // ---- 3) compute: 2 voxels / thread ----
  const float OM = (float)(2.0 * 3.14159265358979323846 * 297200000.0 *
                           1.256637061e-06);  // omega * mu0
  const float c  = 0.5f;                      // 1/(2*dx)
  const int col = tid & 127;
  const int r0  = tid >> 7;  // 0..1

  const size_t nb = (size_t)n * 12;

#define EL(ch, rr, cc) els[((ch) * ROWS + (rr)) * RSTR + 4 + (cc)]

  for (int it = 0; it < HT / 2; ++it) {
    const int hl = r0 + 2 * it;  // 0..3
    const int h  = h0 + hl;
    const int lr = hl + 1;       // staged row index

    // plane-d stencil taps from LDS (zero halos handle h/w boundaries)
    const float ex_w_m = EL(0, lr, col - 1), ex_w_p = EL(0, lr, col + 1);
    const float ex_h_m = EL(0, lr - 1, col), ex_h_p = EL(0, lr + 1, col);
    const float ey_w_m = EL(1, lr, col - 1), ey_w_p = EL(1, lr, col + 1);
    const float ez_h_m = EL(2, lr - 1, col), ez_h_p = EL(2, lr + 1, col);
    const float fx_w_m = EL(3, lr, col - 1), fx_w_p = EL(3, lr, col + 1);
    const float fx_h_m = EL(3, lr - 1, col), fx_h_p = EL(3, lr + 1, col);
    const float fy_w_m = EL(4, lr, col - 1), fy_w_p = EL(4, lr, col + 1);
    const float fz_h_m = EL(5, lr - 1, col), fz_h_p = EL(5, lr + 1, col);

    // d+-1 taps straight from global, coalesced b32 (reused by neighbor
    // d-tiles via L2 -> keep regular-temporal)
    const size_t sp = (size_t)h * DIM + col;
    float ey_d_m = 0.f, ez_d_m = 0.f, fy_d_m = 0.f, fz_d_m = 0.f;
    float ey_d_p = 0.f, ez_d_p = 0.f, fy_d_p = 0.f, fz_d_p = 0.f;
    if (d > 0) {
      const size_t b = (size_t)(d - 1) * PLANE + sp;
      ey_d_m = f[(nb + 1) * VOL + b];
      ez_d_m = f[(nb + 2) * VOL + b];
      fy_d_m = f[(nb + 4) * VOL + b];
      fz_d_m = f[(nb + 5) * VOL + b];
    }
    if (d < DIM - 1) {
      const size_t b = (size_t)(d + 1) * PLANE + sp;
      ey_d_p = f[(nb + 1) * VOL + b];
      ez_d_p = f[(nb + 2) * VOL + b];
      fy_d_p = f[(nb + 4) * VOL + b];
      fz_d_p = f[(nb + 5) * VOL + b];
    }
    // H fields at (d,h,w): read exactly once ever -> non-temporal
    const size_t bc = (size_t)d * PLANE + sp;
    const float hrx = __builtin_nontemporal_load(&f[(nb + 6) * VOL + bc]);
    const float hry = __builtin_nontemporal_load(&f[(nb + 7) * VOL + bc]);
    const float hrz = __builtin_nontemporal_load(&f[(nb + 8) * VOL + bc]);
    const float hix = __builtin_nontemporal_load(&f[(nb + 9) * VOL + bc]);
    const float hiy = __builtin_nontemporal_load(&f[(nb + 10) * VOL + bc]);
    const float hiz = __builtin_nontemporal_load(&f[(nb + 11) * VOL + bc]);

    // curl(E) with axes x=d, y=h, z=w (matches reference filter taps)
    const float crx = c * ((ez_h_p - ez_h_m) - (ey_w_p - ey_w_m));
    const float cry = c * ((ex_w_p - ex_w_m) - (ez_d_p - ez_d_m));
    const float crz = c * ((ey_d_p - ey_d_m) - (ex_h_p - ex_h_m));
    const float cix = c * ((fz_h_p - fz_h_m) - (fy_w_p - fy_w_m));
    const float ciy = c * ((fx_w_p - fx_w_m) - (fz_d_p - fz_d_m));
    const float ciz = c * ((fy_d_p - fy_d_m) - (fx_h_p - fx_h_m));

    const float rrx = crx - OM * hix;
    const float rry = cry - OM * hiy;
    const float rrz = crz - OM * hiz;
    const float rix = cix + OM * hrx;
    const float riy = ciy + OM * hry;
    const float riz = ciz + OM * hrz;

    const float loss = (rrx * rrx + rry * rry + rrz * rrz +
                        rix * rix + riy * riy + riz * riz) * (1.0f / 3.0f);

    // written once, never re-read -> non-temporal store
    __builtin_nontemporal_store(loss, &out[((size_t)n * DIM + d) * PLANE + sp]);
  }
#undef EL
}

extern "C" void kernel_launch(void* const* d_in, const int* in_sizes, int n_in,
                              void* d_out, int out_size, void* d_ws, size_t ws_size,
                              hipStream_t stream) {
  (void)in_sizes; (void)n_in; (void)d_ws; (void)ws_size; (void)out_size;
  const float* field = (const float*)d_in[0];
  float* out = (float*)d_out;
  // 2 batches * 128 planes * 32 h-tiles
  faraday_loss_kernel<<<NB * DIM * (DIM / HT), 256, 0, stream>>>(field, out);
}